// ReconstructionLoss_7808250544424
// MI455X (gfx1250) — compile-verified
//
#include <hip/hip_runtime.h>

typedef float v2f __attribute__((ext_vector_type(2)));
typedef float v4f __attribute__((ext_vector_type(4)));
typedef float v8f __attribute__((ext_vector_type(8)));

// For output pixel (b, r, c): k = (r%4)*4 + (c%4) selects the y source:
//   y[b, kCH[k], 4*(r/4)+kRO[k], 4*(c/4)+kCO[k]]
// Derived from rr/gr/gb/bb stacking + _PERM (channel 15 duplicates channel 3's
// source per _PERM[15]==5; y channel 11 is never read).
__constant__ int kCH[16] = {0, 3, 1, 4,   6, 9, 7, 10,   1, 4, 2, 5,   8, 10, 9, 4};
__constant__ int kRO[16] = {0, 0, 0, 0,   1, 1, 1, 1,    2, 2, 2, 2,   3, 3, 3, 0};
__constant__ int kCO[16] = {0, 1, 2, 3,   0, 1, 2, 3,    0, 1, 2, 3,   0, 1, 2, 3};

// Exact fp32 full-wave (32-lane) sum using V_WMMA_F32_16X16X4_F32.
// A(16x4): lane L<16 holds A[L][0]=v, A[L][1]=0 ; lane L>=16 holds A[L-16][2]=v, A[L-16][3]=0.
// B(4x16) = all ones  ->  D[m][n] = v[m] + v[m+16]  (independent of n, any B layout works).
// Per-lane sum of the 8 D VGPRs gives half-wave sums; one shfl_xor(16) completes it.
__device__ __forceinline__ float wave_reduce_wmma(float v) {
    v2f a; a[0] = v;    a[1] = 0.0f;
    v2f b; b[0] = 1.0f; b[1] = 1.0f;
    v8f c = {};
    v8f d = __builtin_amdgcn_wmma_f32_16x16x4_f32(
        /*neg_a=*/false, a, /*neg_b=*/false, b,
        /*c_mod=*/(short)0, c, /*reuse_a=*/false, /*reuse_b=*/false);
    float s = d[0] + d[1] + d[2] + d[3] + d[4] + d[5] + d[6] + d[7];
    s += __shfl_xor(s, 16, 32);
    return s;
}

__device__ __forceinline__ float block_reduce(float acc) {
    float ws = wave_reduce_wmma(acc);          // uniform: all lanes execute, EXEC all-1s
    __shared__ float sm[8];
    const int wid  = threadIdx.x >> 5;
    const int lane = threadIdx.x & 31;
    if (lane == 0) sm[wid] = ws;
    __syncthreads();
    float t = 0.0f;
    if (threadIdx.x == 0) {
        #pragma unroll
        for (int w = 0; w < 8; ++w) t += sm[w];
    }
    return t;                                   // valid in thread 0 only
}

// Stage 1: grid-stride over 2M column-groups (4 output pixels each).
// x read as NT float4 stream; 4 scattered y loads per group via tables.
__global__ __launch_bounds__(256) void loss_partial_kernel(
    const float* __restrict__ x, const float* __restrict__ y,
    float* __restrict__ part, int ngroups) {

    float acc = 0.0f;
    const int stride = gridDim.x * blockDim.x;
    for (int idx = blockIdx.x * blockDim.x + threadIdx.x; idx < ngroups; idx += stride) {
        const int j = idx & 255;          // column group: c = 4j + q
        const int r = (idx >> 8) & 1023;  // output row
        const int b = idx >> 18;          // batch
        const int p = r & 3;              // wave-uniform (32 lanes share one row)
        const int i = r >> 2;

        const size_t xoff = ((((size_t)b << 10) + (size_t)r) << 10) + ((size_t)j << 2);
        const v4f xv = __builtin_nontemporal_load(
            reinterpret_cast<const v4f*>(x + xoff));
        const float xs[4] = {xv[0], xv[1], xv[2], xv[3]};

        #pragma unroll
        for (int q = 0; q < 4; ++q) {
            const int k  = (p << 2) | q;
            const int ch = kCH[k];
            const int ro = kRO[k];
            const int co = kCO[k];
            const size_t yoff =
                ((((size_t)(b * 12 + ch) << 10) + (size_t)((i << 2) + ro)) << 10)
                + ((size_t)j << 2) + (size_t)co;
            acc += fabsf(y[yoff] - xs[q]);
        }
    }

    const float t = block_reduce(acc);
    if (threadIdx.x == 0) part[blockIdx.x] = t;
}

// Stage 2: single block folds the per-block partials, scales to the mean.
__global__ __launch_bounds__(256) void loss_final_kernel(
    const float* __restrict__ part, int n, float* __restrict__ out, float invN) {
    float acc = 0.0f;
    for (int i = threadIdx.x; i < n; i += 256) acc += part[i];
    const float t = block_reduce(acc);
    if (threadIdx.x == 0) out[0] = t * invN;
}

extern "C" void kernel_launch(void* const* d_in, const int* in_sizes, int n_in,
                              void* d_out, int out_size, void* d_ws, size_t ws_size,
                              hipStream_t stream) {
    const float* x = (const float*)d_in[0];   // (8,1,1024,1024) f32
    const float* y = (const float*)d_in[1];   // (8,12,1024,1024) f32
    float* part = (float*)d_ws;

    int nblocks = 2048;                       // 8 KB of workspace partials
    if (ws_size < (size_t)nblocks * sizeof(float)) {
        nblocks = (int)(ws_size / sizeof(float));
        if (nblocks < 1) nblocks = 1;
    }
    const int ngroups = 8 * 1024 * 256;       // 2,097,152 column-groups

    loss_partial_kernel<<<nblocks, 256, 0, stream>>>(x, y, part, ngroups);
    loss_final_kernel<<<1, 256, 0, stream>>>(part, nblocks, (float*)d_out,
                                             1.0f / 8388608.0f);
}